// NeuronalDynamics_79413945303718
// MI455X (gfx1250) — compile-verified
//
#include <hip/hip_runtime.h>
#include <stdint.h>

// Problem constants (match reference)
#define MCOLS 64          // state columns
#define UTHR 3.5f
#define DSLOPE 2.0f

// Tiling
#define CHUNK 512         // edges per wave-chunk (DMA'd by TDM into LDS)
#define WAVES_PER_BLOCK 8 // 256 threads, 8 wave32s

typedef uint32_t u32x4 __attribute__((ext_vector_type(4)));
typedef int      i32x8 __attribute__((ext_vector_type(8)));
typedef int      i32x4 __attribute__((ext_vector_type(4)));

// ---- Tensor DMA descriptor builders (D#) -----------------------------------
// Group 0 (128b): [1:0]=count=1, [63:32]=lds_addr, [120:64]=global_addr,
//                 [127:126]=type=2 ("image")
__device__ __forceinline__ u32x4 tdm_g0(uint32_t lds_addr, uint64_t gaddr) {
    u32x4 g;
    g.x = 1u;                                   // count=1, user-mode, no gather
    g.y = lds_addr;                             // LDS byte address
    g.z = (uint32_t)(gaddr & 0xFFFFFFFFu);      // global_addr[31:0]
    g.w = (uint32_t)((gaddr >> 32) & 0x01FFFFFFu) | (2u << 30); // [56:32] | type=2
    return g;
}

// Group 1 (256b): 1-D tile of 4-byte elements.
//   data_size=2 (4B) @ [17:16]; tensor_dim0=valid @ [79:48];
//   tensor_dim1=1 @ [111:80]; tile_dim0=CHUNK @ [127:112]; tile_dim1=1 @ [143:128]
//   tensor_dim0_stride=CHUNK @ [207:160] (unused for a single-row tile)
__device__ __forceinline__ i32x8 tdm_g1(uint32_t valid) {
    i32x8 g = {0, 0, 0, 0, 0, 0, 0, 0};
    g[0] = (int)(2u << 16);                        // data_size = 4 bytes
    g[1] = (int)((valid & 0xFFFFu) << 16);         // tensor_dim0[15:0]
    g[2] = (int)(((valid >> 16) & 0xFFFFu) | (1u << 16)); // tensor_dim0[31:16] | tensor_dim1=1
    g[3] = (int)((uint32_t)CHUNK << 16);           // tile_dim0 = CHUNK
    g[4] = 1;                                      // tile_dim1 = 1
    g[5] = CHUNK;                                  // tensor_dim0_stride (don't-care)
    return g;
}

// ---- Kernel 1: s = sigmoid(D*x - U); out = -x ------------------------------
__global__ void __launch_bounds__(256)
nd_sigmoid_init(const float* __restrict__ x, float* __restrict__ s,
                float* __restrict__ out, int n) {
    int i4 = (blockIdx.x * blockDim.x + threadIdx.x) * 4;
    if (i4 + 3 < n) {
        float4 xv = *(const float4*)(x + i4);
        float4 sv, ov;
        sv.x = 1.0f / (1.0f + __expf(UTHR - DSLOPE * xv.x));
        sv.y = 1.0f / (1.0f + __expf(UTHR - DSLOPE * xv.y));
        sv.z = 1.0f / (1.0f + __expf(UTHR - DSLOPE * xv.z));
        sv.w = 1.0f / (1.0f + __expf(UTHR - DSLOPE * xv.w));
        ov.x = -xv.x; ov.y = -xv.y; ov.z = -xv.z; ov.w = -xv.w;
        *(float4*)(s + i4) = sv;
        *(float4*)(out + i4) = ov;
    } else {
        for (int i = i4; i < n; ++i) {
            float xs = x[i];
            s[i]   = 1.0f / (1.0f + __expf(UTHR - DSLOPE * xs));
            out[i] = -xs;
        }
    }
}

// ---- Kernel 2: edge-parallel SpMM scatter-add ------------------------------
// One wave per CHUNK edges. TDM streams row/col/w into LDS (TENSORcnt);
// lanes cover the 64 columns (2 floats/lane): 256B coalesced gather from s
// (L2-resident) and 2 coalesced global_atomic_add_f32 per lane per edge.
__global__ void __launch_bounds__(WAVES_PER_BLOCK * 32)
nd_edge_spmm(const int* __restrict__ row, const int* __restrict__ col,
             const float* __restrict__ ew, const float* __restrict__ s,
             float* __restrict__ out, int nEdges) {
    __shared__ int   lds_row[WAVES_PER_BLOCK][CHUNK];
    __shared__ int   lds_col[WAVES_PER_BLOCK][CHUNK];
    __shared__ float lds_w  [WAVES_PER_BLOCK][CHUNK];

    const int lane = (int)(threadIdx.x & 31u);
    const int wv   = __builtin_amdgcn_readfirstlane((int)(threadIdx.x >> 5));
    const int chunkId =
        __builtin_amdgcn_readfirstlane((int)(blockIdx.x * WAVES_PER_BLOCK) + wv);
    const int base = chunkId * CHUNK;

    if (base < nEdges) {
        const int      rem   = nEdges - base;
        const uint32_t valid = (uint32_t)(rem < CHUNK ? rem : CHUNK);

        const uint32_t lr = (uint32_t)(uintptr_t)&lds_row[wv][0];
        const uint32_t lc = (uint32_t)(uintptr_t)&lds_col[wv][0];
        const uint32_t lw = (uint32_t)(uintptr_t)&lds_w[wv][0];

        const i32x8 g1  = tdm_g1(valid);
        const i32x4 gz4 = {0, 0, 0, 0};
        const i32x8 gz8 = {0, 0, 0, 0, 0, 0, 0, 0};

        // Three async DMA tiles: row, col, w  (TENSORcnt += 3)
        __builtin_amdgcn_tensor_load_to_lds(
            tdm_g0(lr, (uint64_t)(uintptr_t)(row + base)), g1, gz4, gz4, gz8, 0);
        __builtin_amdgcn_tensor_load_to_lds(
            tdm_g0(lc, (uint64_t)(uintptr_t)(col + base)), g1, gz4, gz4, gz8, 0);
        __builtin_amdgcn_tensor_load_to_lds(
            tdm_g0(lw, (uint64_t)(uintptr_t)(ew + base)), g1, gz4, gz4, gz8, 0);
        __builtin_amdgcn_s_wait_tensorcnt(0);

        const int col0 = lane * 2;
        const uint32_t vmain = valid & ~7u;

        for (uint32_t e0 = 0; e0 < vmain; e0 += 8) {
            int r[8]; float w8[8]; float2 sv[8];
            #pragma unroll
            for (int j = 0; j < 8; ++j) {       // 8 gathers in flight
                const int c = lds_col[wv][e0 + j];
                r[j]  = lds_row[wv][e0 + j];
                w8[j] = lds_w[wv][e0 + j];
                sv[j] = *(const float2*)(s + (size_t)c * MCOLS + col0);
            }
            #pragma unroll
            for (int j = 0; j < 8; ++j) {       // 16 coalesced f32 atomics
                float* dst = out + (size_t)r[j] * MCOLS + col0;
                atomicAdd(dst + 0, w8[j] * sv[j].x);
                atomicAdd(dst + 1, w8[j] * sv[j].y);
            }
        }
        for (uint32_t e = vmain; e < valid; ++e) {   // tail (last chunk only)
            const int   c  = lds_col[wv][e];
            const int   r  = lds_row[wv][e];
            const float wj = lds_w[wv][e];
            float2 sve = *(const float2*)(s + (size_t)c * MCOLS + col0);
            float* dst = out + (size_t)r * MCOLS + col0;
            atomicAdd(dst + 0, wj * sve.x);
            atomicAdd(dst + 1, wj * sve.y);
        }
    }
}

// ---- Launch ----------------------------------------------------------------
extern "C" void kernel_launch(void* const* d_in, const int* in_sizes, int n_in,
                              void* d_out, int out_size, void* d_ws, size_t ws_size,
                              hipStream_t stream) {
    // setup_inputs order: t, x, row, col, edge_w
    const float* x   = (const float*)d_in[1];
    const int*   row = (const int*)d_in[2];
    const int*   col = (const int*)d_in[3];
    const float* ew  = (const float*)d_in[4];
    const int nx = in_sizes[1];   // N*M
    const int nE = in_sizes[2];   // E

    float* out = (float*)d_out;
    float* s   = (float*)d_ws;    // N*M floats of scratch for sigmoid state

    // Kernel 1: sigmoid + out = -x
    {
        const int threads = 256;
        const int blocks = (nx / 4 + threads - 1) / threads;
        nd_sigmoid_init<<<blocks, threads, 0, stream>>>(x, s, out, nx);
    }
    // Kernel 2: TDM-staged edge scatter-add
    {
        const int chunks = (nE + CHUNK - 1) / CHUNK;
        const int blocks = (chunks + WAVES_PER_BLOCK - 1) / WAVES_PER_BLOCK;
        nd_edge_spmm<<<blocks, WAVES_PER_BLOCK * 32, 0, stream>>>(
            row, col, ew, s, out, nE);
    }
}